// ProgressiveDropout_86569360818324
// MI455X (gfx1250) — compile-verified
//
#include <hip/hip_runtime.h>

// out[i, j] = x[i, j] if j < samples[i] else 0
// B = 8192 rows, C = 4096 cols, fp32. Pure streaming op:
//   write 128 MiB always; read only the kept prefix (~64 MiB expected).
// One block per row, 256 threads (8 wave32), 128-bit non-temporal VMEM.
//
// The keep/drop decision is made *per wave* with scalar branches:
// s is block-uniform and a wave covers a contiguous 128-channel span,
// so comparing s against readfirstlane(j) gives s_cmp/s_cbranch control
// flow with no EXEC-mask churn. Only the one wave straddling the cutoff
// takes the per-element select path.

typedef float v4f __attribute__((ext_vector_type(4)));

__device__ __forceinline__ void process_chunk(const v4f* __restrict__ xin,
                                              v4f* __restrict__ xout,
                                              int v, int s) {
  const int j  = v << 2;                              // this lane's first channel
  const int jw = __builtin_amdgcn_readfirstlane(j);   // wave's first channel
  // This wave covers channels [jw, jw + 128).
  if (s >= jw + 128) {
    // Whole wave kept: NT copy (global_load_b128 / global_store_b128, TH=NT)
    v4f val = __builtin_nontemporal_load(&xin[v]);
    __builtin_nontemporal_store(val, &xout[v]);
  } else if (s <= jw) {
    // Whole wave dropped: no read, store zeros.
    v4f z = {0.f, 0.f, 0.f, 0.f};
    __builtin_nontemporal_store(z, &xout[v]);
  } else {
    // Straddling wave (one per row): per-element select.
    v4f val = __builtin_nontemporal_load(&xin[v]);
    val.x = (j + 0 < s) ? val.x : 0.f;
    val.y = (j + 1 < s) ? val.y : 0.f;
    val.z = (j + 2 < s) ? val.z : 0.f;
    val.w = (j + 3 < s) ? val.w : 0.f;
    __builtin_nontemporal_store(val, &xout[v]);
  }
}

// ITERS = (C/4) / 256 chunk-iterations per thread, known at compile time.
template <int ITERS>
__global__ __launch_bounds__(256) void progressive_dropout_fixed(
    const float* __restrict__ x,
    const int* __restrict__ samples,
    float* __restrict__ out) {
  const int row = blockIdx.x;
  const int s = samples[row];  // block-uniform -> s_load_b32

  const long long base = (long long)row * (long long)(ITERS * 256 * 4);
  const v4f* __restrict__ xin  = reinterpret_cast<const v4f*>(x + base);
  v4f* __restrict__       xout = reinterpret_cast<v4f*>(out + base);

  const int tid = threadIdx.x;
#pragma unroll
  for (int k = 0; k < ITERS; ++k) {
    process_chunk(xin, xout, tid + k * 256, s);
  }
}

// Generic fallback for unexpected shapes (C must be a multiple of 4).
__global__ __launch_bounds__(256) void progressive_dropout_generic(
    const float* __restrict__ x,
    const int* __restrict__ samples,
    float* __restrict__ out,
    int C) {
  const int row = blockIdx.x;
  const int s = samples[row];

  const long long base = (long long)row * (long long)C;
  const v4f* __restrict__ xin  = reinterpret_cast<const v4f*>(x + base);
  v4f* __restrict__       xout = reinterpret_cast<v4f*>(out + base);

  const int nvec = C >> 2;
  for (int v = threadIdx.x; v < nvec; v += 256) {
    process_chunk(xin, xout, v, s);
  }
}

extern "C" void kernel_launch(void* const* d_in, const int* in_sizes, int n_in,
                              void* d_out, int out_size, void* d_ws, size_t ws_size,
                              hipStream_t stream) {
  (void)n_in; (void)d_ws; (void)ws_size; (void)out_size;
  const float* x       = (const float*)d_in[0];
  const int*   samples = (const int*)d_in[1];
  float*       out     = (float*)d_out;

  const int B = in_sizes[1];        // 8192 rows (one sample per row)
  const int C = in_sizes[0] / B;    // 4096 cols

  dim3 grid(B);
  dim3 block(256);
  if (C == 4096) {
    // 4096 floats/row = 1024 float4 chunks = 4 per thread.
    progressive_dropout_fixed<4><<<grid, block, 0, stream>>>(x, samples, out);
  } else {
    progressive_dropout_generic<<<grid, block, 0, stream>>>(x, samples, out, C);
  }
}